// AttentionMechanism_41369124995539
// MI455X (gfx1250) — compile-verified
//
#include <hip/hip_runtime.h>
#include <math.h>

// ---------------------------------------------------------------------------
// AttentionMechanism on MI455X (gfx1250, wave32)
//
// scores[b,s] = <CV[b,:,s], u[b]> + const(b), u[b] = W^T h[b]; the constant
// (bias . h) cancels under softmax, so bias is dead. The workload is pure
// bandwidth: CV = 268 MB (one flash-style pass), attentions out = 67 MB.
//   K1 (WMMA f32 16x16x4): u = W^T h                 [256x256]x[256x32]
//   K2 (flash chunks):     per (b, 1024-col chunk): b128 score pass ->
//                          chunk max/sumexp -> NT b128 weighted-ctx pass
//                          (chunk re-read hits L2 -> single HBM pass)
//   K3 (combine):          global softmax rescale, write 64 context copies
//   K4 (attn writer):      attn = exp(score-M)/L, 64 NT b128 broadcast copies
// ---------------------------------------------------------------------------

typedef __attribute__((ext_vector_type(2))) float v2f;
typedef __attribute__((ext_vector_type(4))) float v4f;
typedef __attribute__((ext_vector_type(8))) float v8f;
typedef __attribute__((ext_vector_type(4))) unsigned char v4u8;

#define BB 32
#define SS 8192
#define DD 256
#define SEQ 64
#define NCHUNK 8
#define CHUNK 1024  // s-columns per workgroup in K2 (4 per thread)

// ---------------------------------------------------------------- Kernel 1 --
// u[b][d] = sum_e W[e][d] * hidden[b][e]   via V_WMMA_F32_16X16X4_F32.
// One wave per 16x16 output tile; 16 d-tiles x 2 b-tiles = 32 waves.
__global__ __launch_bounds__(32) void proj_wmma(const float* __restrict__ W,
                                                const float* __restrict__ hidden,
                                                float* __restrict__ u) {
  const int tile = blockIdx.x;         // 0..31
  const int dt = tile & 15;            // d-tile (M)
  const int bt = tile >> 4;            // batch-tile (N)
  const int lane = threadIdx.x;        // 0..31
  const int mn = lane & 15;            // M index for A, N index for B
  const int khalf = (lane >> 4) << 1;  // lanes 0-15: K+{0,1}; 16-31: K+{2,3}

  v8f c = {};
  for (int e = 0; e < DD; e += 4) {
    v2f a, bm;
    // A[m, k] = W[k][dt*16+m]  (A = W^T), ISA f32 16x4 layout
    a.x = W[(e + khalf + 0) * DD + dt * 16 + mn];
    a.y = W[(e + khalf + 1) * DD + dt * 16 + mn];
    // B[k, n] = hidden[bt*16+n][k], ISA f32 4x16 layout
    bm.x = hidden[(bt * 16 + mn) * DD + e + khalf + 0];
    bm.y = hidden[(bt * 16 + mn) * DD + e + khalf + 1];
    c = __builtin_amdgcn_wmma_f32_16x16x4_f32(
        /*neg_a=*/false, a, /*neg_b=*/false, bm,
        /*c_mod=*/(short)0, c, /*reuse_a=*/false, /*reuse_b=*/false);
  }
  // C/D layout: VGPR r, lanes 0-15 -> M=r, lanes 16-31 -> M=r+8; N = lane&15
  const int mbase = (lane >> 4) * 8;
#pragma unroll
  for (int r = 0; r < 8; ++r) {
    u[(bt * 16 + mn) * DD + dt * 16 + mbase + r] = c[r];
  }
}

// ---------------------------------------------------------------- Kernel 2 --
// One workgroup per (chunk, b). 256 threads, 4 columns per thread (b128).
__global__ __launch_bounds__(256) void scores_flash(
    const float* __restrict__ cv, const unsigned char* __restrict__ mask,
    const float* __restrict__ u, float* __restrict__ scores,
    float* __restrict__ mpart, float* __restrict__ lpart,
    float* __restrict__ ctxpart) {
  __shared__ float u_lds[DD];
  __shared__ float w_lds[CHUNK];
  __shared__ float red[256];

  const int t = threadIdx.x;
  const int chunk = blockIdx.x;
  const int b = blockIdx.y;
  const int s0 = chunk * CHUNK;

  u_lds[t] = u[b * DD + t];
  __syncthreads();

  // ---- phase 1: 4 scores per thread, b128 loads --------------------------
  const size_t cvb = (size_t)b * DD * SS;
  const v4f* col4 = reinterpret_cast<const v4f*>(cv + cvb + s0) + t;
  v4f acc = {0.f, 0.f, 0.f, 0.f};
#pragma unroll 4
  for (int d = 0; d < DD; ++d) {
    const float uv = u_lds[d];
    const v4f v = col4[(size_t)d * (SS / 4)];
    acc.x = fmaf(v.x, uv, acc.x);
    acc.y = fmaf(v.y, uv, acc.y);
    acc.z = fmaf(v.z, uv, acc.z);
    acc.w = fmaf(v.w, uv, acc.w);
  }
  const v4u8 mk =
      *reinterpret_cast<const v4u8*>(mask + (size_t)b * SS + s0 + 4 * t);
  if (mk.x) acc.x = -INFINITY;
  if (mk.y) acc.y = -INFINITY;
  if (mk.z) acc.z = -INFINITY;
  if (mk.w) acc.w = -INFINITY;
  *reinterpret_cast<v4f*>(scores + (size_t)b * SS + s0 + 4 * t) = acc;

  // ---- phase 2: chunk max + sumexp (fixed-order tree, deterministic) -----
  red[t] = fmaxf(fmaxf(acc.x, acc.y), fmaxf(acc.z, acc.w));
  __syncthreads();
  for (int off = 128; off > 0; off >>= 1) {
    if (t < off) red[t] = fmaxf(red[t], red[t + off]);
    __syncthreads();
  }
  const float m_c = red[0];
  __syncthreads();

  v4f w;
  w.x = mk.x ? 0.f : expf(acc.x - m_c);
  w.y = mk.y ? 0.f : expf(acc.y - m_c);
  w.z = mk.z ? 0.f : expf(acc.z - m_c);
  w.w = mk.w ? 0.f : expf(acc.w - m_c);
  *reinterpret_cast<v4f*>(&w_lds[4 * t]) = w;
  red[t] = (w.x + w.y) + (w.z + w.w);
  __syncthreads();
  for (int off = 128; off > 0; off >>= 1) {
    if (t < off) red[t] += red[t + off];
    __syncthreads();
  }
  const float l_c = red[0];

  // ---- phase 3: chunk-weighted context; last-use NT reads (L2 hot) -------
  float a3 = 0.f;
  const v4f* row4 = reinterpret_cast<const v4f*>(cv + cvb + (size_t)t * SS + s0);
#pragma unroll 4
  for (int i = 0; i < CHUNK / 4; ++i) {
    const v4f v = __builtin_nontemporal_load(row4 + i);
    a3 = fmaf(v.x, w_lds[4 * i + 0], a3);
    a3 = fmaf(v.y, w_lds[4 * i + 1], a3);
    a3 = fmaf(v.z, w_lds[4 * i + 2], a3);
    a3 = fmaf(v.w, w_lds[4 * i + 3], a3);
  }
  ctxpart[((size_t)b * NCHUNK + chunk) * DD + t] = a3;
  if (t == 0) {
    mpart[b * NCHUNK + chunk] = m_c;
    lpart[b * NCHUNK + chunk] = l_c;
  }
}

// ---------------------------------------------------------------- Kernel 3 --
__global__ __launch_bounds__(256) void combine(
    const float* __restrict__ mpart, const float* __restrict__ lpart,
    const float* __restrict__ ctxpart, float* __restrict__ out_ctx,
    float* __restrict__ Mb, float* __restrict__ invLb) {
  const int b = blockIdx.x;
  const int t = threadIdx.x;

  float M = -INFINITY;
#pragma unroll
  for (int c = 0; c < NCHUNK; ++c) M = fmaxf(M, mpart[b * NCHUNK + c]);
  float L = 0.f;
  float sc[NCHUNK];
#pragma unroll
  for (int c = 0; c < NCHUNK; ++c) {
    sc[c] = expf(mpart[b * NCHUNK + c] - M);
    L += lpart[b * NCHUNK + c] * sc[c];
  }
  const float invL = 1.0f / L;

  float ctx = 0.f;
#pragma unroll
  for (int c = 0; c < NCHUNK; ++c)
    ctx = fmaf(ctxpart[((size_t)b * NCHUNK + c) * DD + t], sc[c], ctx);
  ctx *= invL;

#pragma unroll 4
  for (int i = 0; i < SEQ; ++i)
    __builtin_nontemporal_store(ctx, &out_ctx[(size_t)i * BB * DD + b * DD + t]);

  if (t == 0) {
    Mb[b] = M;
    invLb[b] = invL;
  }
}

// ---------------------------------------------------------------- Kernel 4 --
// 4 attention values per thread; 64 broadcast copies via NT b128 stores.
__global__ __launch_bounds__(256) void attn_write(
    const float* __restrict__ scores, const float* __restrict__ Mb,
    const float* __restrict__ invLb, float* __restrict__ out_attn) {
  const int e0 = (blockIdx.x * 256 + threadIdx.x) * 4;  // 0 .. B*S-1, step 4
  const int b = e0 >> 13;  // 4-groups never straddle a batch (8192 % 4 == 0)
  const float M = Mb[b];
  const float invL = invLb[b];
  const v4f s = *reinterpret_cast<const v4f*>(scores + e0);
  v4f a;
  a.x = (s.x == -INFINITY) ? 0.f : expf(s.x - M) * invL;
  a.y = (s.y == -INFINITY) ? 0.f : expf(s.y - M) * invL;
  a.z = (s.z == -INFINITY) ? 0.f : expf(s.z - M) * invL;
  a.w = (s.w == -INFINITY) ? 0.f : expf(s.w - M) * invL;
#pragma unroll 4
  for (int i = 0; i < SEQ; ++i)
    __builtin_nontemporal_store(
        a, reinterpret_cast<v4f*>(out_attn + (size_t)i * BB * SS + e0));
}

// --------------------------------------------------------------------------
extern "C" void kernel_launch(void* const* d_in, const int* in_sizes, int n_in,
                              void* d_out, int out_size, void* d_ws,
                              size_t ws_size, hipStream_t stream) {
  // inputs: [0]=seqlen(int)=64, [1]=hidden[1,B,D] f32, [2]=contextvects[B,D,S]
  //         f32, [3]=W[D,D] f32, [4]=bias[D] f32 (cancels under softmax),
  //         [5]=padding_mask[B,S,1] bool (1 byte/elem)
  const float* hidden = (const float*)d_in[1];
  const float* cv = (const float*)d_in[2];
  const float* W = (const float*)d_in[3];
  const unsigned char* mask = (const unsigned char*)d_in[5];

  float* ws = (float*)d_ws;
  float* u = ws;                         // B*D        =   8192 f
  float* scores = u + BB * DD;           // B*S        = 262144 f
  float* mpart = scores + BB * SS;       // B*NCHUNK   =    256 f
  float* lpart = mpart + BB * NCHUNK;    // B*NCHUNK   =    256 f
  float* ctxpart = lpart + BB * NCHUNK;  // B*NCHUNK*D =  65536 f
  float* Mb = ctxpart + (size_t)BB * NCHUNK * DD;  // B =    32 f
  float* invLb = Mb + BB;                          // B =    32 f
  // total ~1.35 MB of d_ws

  float* out_ctx = (float*)d_out;                     // [SEQ,B,D]
  float* out_attn = out_ctx + (size_t)SEQ * BB * DD;  // [SEQ,B,S]

  proj_wmma<<<32, 32, 0, stream>>>(W, hidden, u);
  scores_flash<<<dim3(NCHUNK, BB), 256, 0, stream>>>(cv, mask, u, scores, mpart,
                                                     lpart, ctxpart);
  combine<<<BB, 256, 0, stream>>>(mpart, lpart, ctxpart, out_ctx, Mb, invLb);
  attn_write<<<(BB * SS) / 1024, 256, 0, stream>>>(scores, Mb, invLb, out_attn);
}